// PeriodicityTransform_90632399880408
// MI455X (gfx1250) — compile-verified
//
#include <hip/hip_runtime.h>
#include <math.h>

// Problem constants (from reference): B=32, T=2048, N=128 -> BN=4096 series.
#define BN_TOTAL 4096
#define T_LEN    2048
#define N_DIM    128
#define K_PER    5
#define PMAX_P   512

typedef float v2f __attribute__((ext_vector_type(2)));
typedef float v8f __attribute__((ext_vector_type(8)));

#define TWO_PI_F 6.28318530717958647692f

#if __has_builtin(__builtin_amdgcn_wmma_f32_16x16x4_f32)
#define HAVE_WMMA_F32X4 1
#endif
#if __has_builtin(__builtin_amdgcn_global_load_async_to_lds_b32)
#define HAVE_ASYNC_LDS 1
#endif

typedef __attribute__((address_space(1))) int* gptr_i32;
typedef __attribute__((address_space(3))) int* lptr_i32;

// One workgroup per series. 256 threads = 8 wave32.
__global__ __launch_bounds__(256) void
periodicity_kernel(const float* __restrict__ x, float* __restrict__ out) {
    __shared__ float ldsSeq[T_LEN];       // series, natural time order
    __shared__ float ldsYr[T_LEN];        // [16][128] complex work (real)
    __shared__ float ldsYi[T_LEN];        // [16][128] complex work (imag)
    __shared__ float ldsMag[1025];        // power spectrum bins 0..1024
    __shared__ float ldsCos[256];         // F16 DFT matrix, cos part  [k1][n1]
    __shared__ float ldsMSin[256];        // F16 DFT matrix, -sin part [k1][n1]
    __shared__ float redV[256];
    __shared__ int   redI[256];
    __shared__ int   sTake[K_PER], sStart[K_PER], sPeriod[K_PER], sCycle[K_PER];

    const int tid  = threadIdx.x;
    const int bn   = blockIdx.x;          // series index = b*128 + n
    const int bIdx = bn >> 7;
    const int nIdx = bn & (N_DIM - 1);
    // seq[t] = x[(b*T + t)*N + n]  (x is [B,T,N]; reference transposes to [B,N,T])
    const size_t gbase = (size_t)bIdx * T_LEN * N_DIM + (size_t)nIdx;

    // ---- Stage 0: DFT-16 coefficient matrices into LDS (one entry per thread)
    {
        const int k1 = tid >> 4, n1 = tid & 15;
        const float th = TWO_PI_F * (float)(k1 * n1) * (1.0f / 16.0f);
        ldsCos[tid]  =  __cosf(th);
        ldsMSin[tid] = -__sinf(th);
    }

    // ---- Stage 1: stage series into LDS (async global->LDS when available)
#ifdef HAVE_ASYNC_LDS
    for (int t = tid; t < T_LEN; t += 256) {
        const float* gp = x + gbase + (size_t)t * N_DIM;
        __builtin_amdgcn_global_load_async_to_lds_b32(
            (gptr_i32)(gp),
            (lptr_i32)(&ldsSeq[t]),
            0, 0);
    }
#if __has_builtin(__builtin_amdgcn_s_wait_asynccnt)
    __builtin_amdgcn_s_wait_asynccnt(0);
#else
    asm volatile("s_wait_asynccnt 0" ::: "memory");
#endif
#else
    for (int t = tid; t < T_LEN; t += 256)
        ldsSeq[t] = x[gbase + (size_t)t * N_DIM];
#endif
    __syncthreads();

    // ---- Stage 2: four-step FFT, N = 16 x 128.
    // Step A (WMMA): Y[k1][n2] = sum_{n1} F16[k1][n1] * seq[n1*128 + n2]
    // Each wave owns 16 columns; K=16 done as 4 chained 16x16x4 f32 WMMAs,
    // twice (cos and -sin) for the complex DFT matrix.
    const int lane  = tid & 31;
    const int wave  = tid >> 5;
    const int col   = wave * 16 + (lane & 15);   // n2 column this lane owns (B/C/D: N = lane%16)
    const int mrow  = lane & 15;                 // A layout: M = lane%16
    const int kbase = (lane < 16) ? 0 : 2;       // A/B: K pair (0,1) vs (2,3)
    const int mofs  = (lane < 16) ? 0 : 8;       // C/D: rows v / v+8

    v8f accR = {};
    v8f accI = {};
#ifdef HAVE_WMMA_F32X4
    #pragma unroll
    for (int kk = 0; kk < 4; ++kk) {
        const int K0 = kk * 4 + kbase;
        v2f aC, aS, b;
        aC.x = ldsCos [mrow * 16 + K0];  aC.y = ldsCos [mrow * 16 + K0 + 1];
        aS.x = ldsMSin[mrow * 16 + K0];  aS.y = ldsMSin[mrow * 16 + K0 + 1];
        b.x  = ldsSeq[K0 * N_DIM + col];
        b.y  = ldsSeq[(K0 + 1) * N_DIM + col];
        accR = __builtin_amdgcn_wmma_f32_16x16x4_f32(false, aC, false, b,
                                                     (short)0, accR, false, false);
        accI = __builtin_amdgcn_wmma_f32_16x16x4_f32(false, aS, false, b,
                                                     (short)0, accI, false, false);
    }
#else
    // VALU fallback with identical per-lane layout
    #pragma unroll
    for (int v = 0; v < 8; ++v) {
        const int M = v + mofs;
        float sr = 0.f, si = 0.f;
        for (int K = 0; K < 16; ++K) {
            const float xv = ldsSeq[K * N_DIM + col];
            sr += ldsCos [M * 16 + K] * xv;
            si += ldsMSin[M * 16 + K] * xv;
        }
        accR[v] = sr; accI[v] = si;
    }
#endif
    // Step B: twiddle by exp(-2*pi*i * k1*n2 / 2048) and spill to LDS [16][128]
    #pragma unroll
    for (int v = 0; v < 8; ++v) {
        const int M = v + mofs;
        const float ang = -TWO_PI_F * (float)(M * col) * (1.0f / (float)T_LEN);
        const float wr = __cosf(ang), wi = __sinf(ang);
        const float yr = accR[v], yi = accI[v];
        ldsYr[M * N_DIM + col] = yr * wr - yi * wi;
        ldsYi[M * N_DIM + col] = yr * wi + yi * wr;
    }
    __syncthreads();

    // Step C: 16 independent 128-point radix-2 DIF FFTs over n2 (rows of [16][128]).
    for (int m = 128; m >= 2; m >>= 1) {
        const int half = m >> 1;
        const float wstep = -TWO_PI_F / (float)m;
        for (int bidx = tid; bidx < 1024; bidx += 256) {   // 16 rows * 64 butterflies
            const int row = bidx >> 6;
            const int j   = bidx & 63;
            const int blk = j / half;
            const int pos = j - blk * half;
            const int i0  = row * N_DIM + blk * m + pos;
            const int i1  = i0 + half;
            const float ang = wstep * (float)pos;
            const float wr = __cosf(ang), wi = __sinf(ang);
            const float ur = ldsYr[i0], ui = ldsYi[i0];
            const float vr = ldsYr[i1], vi = ldsYi[i1];
            ldsYr[i0] = ur + vr;
            ldsYi[i0] = ui + vi;
            const float dr = ur - vr, di = ui - vi;
            ldsYr[i1] = dr * wr - di * wi;
            ldsYi[i1] = dr * wi + di * wr;
        }
        __syncthreads();
    }

    // ---- Stage 3: power spectrum. X(k1 + 16*k2) sits at [k1][bitrev7(k2)].
    for (int e = tid; e < T_LEN; e += 256) {
        const int k1 = e >> 7;
        const int k2 = e & 127;
        const int k  = k1 + 16 * k2;
        if (k <= 1024) {
            const int p = k1 * N_DIM + (int)(__brev((unsigned)k2) >> 25);
            const float re = ldsYr[p], im = ldsYi[p];
            ldsMag[k] = (k == 0) ? 0.0f : (re * re + im * im);
        }
    }
    __syncthreads();

    // ---- Stage 4: iterative top-5 (lower index wins ties, like lax.top_k)
    for (int p = 0; p < K_PER; ++p) {
        float bv = -1.0f; int bi = 0;
        for (int k = tid; k <= 1024; k += 256) {
            const float v = ldsMag[k];
            if (v > bv) { bv = v; bi = k; }
        }
        redV[tid] = bv; redI[tid] = bi;
        __syncthreads();
        for (int s = 128; s > 0; s >>= 1) {
            if (tid < s) {
                const float ov = redV[tid + s]; const int oi = redI[tid + s];
                if (ov > redV[tid] || (ov == redV[tid] && oi < redI[tid])) {
                    redV[tid] = ov; redI[tid] = oi;
                }
            }
            __syncthreads();
        }
        if (tid == 0) {
            const int idx = redI[0];
            ldsMag[idx] = -2.0f;                 // exclude for next pass
            const int kidx = (idx < 1) ? 1 : idx;
            int period = T_LEN / kidx;
            if (period > PMAX_P) period = PMAX_P;
            if (period < 2)      period = 2;     // MIN_PERIOD_THRESHOLD
            int cycles = T_LEN / period;
            if (cycles < 1) cycles = 1;
            const int take = cycles * period;
            sPeriod[p] = period; sCycle[p] = cycles;
            sTake[p]   = take;   sStart[p] = T_LEN - take;
        }
        __syncthreads();
    }

    // ---- Stage 5: outputs. d_out = [tiles | periods | cycles] flat.
    const size_t periodsBase = (size_t)BN_TOTAL * K_PER * T_LEN;
    const size_t cyclesBase  = periodsBase + (size_t)BN_TOTAL * K_PER;
    if (tid < K_PER) {
        out[periodsBase + (size_t)bn * K_PER + tid] = (float)sPeriod[tid];
        out[cyclesBase  + (size_t)bn * K_PER + tid] = (float)sCycle[tid];
    }
    // Tile rows start at 8KB-aligned offsets -> float4 (global_store_b128) is safe.
    for (int p = 0; p < K_PER; ++p) {
        const size_t base = ((size_t)bn * K_PER + p) * (size_t)T_LEN;
        const int take = sTake[p], start = sStart[p];
        float4* out4 = (float4*)(out + base);
        for (int t4 = tid; t4 < T_LEN / 4; t4 += 256) {
            const int t = t4 * 4;
            float4 v;
            int i0 = start + t;
            i0 = (i0 > T_LEN - 1) ? (T_LEN - 1) : i0;   // clamp speculated LDS reads
            v.x = (t + 0 < take) ? ldsSeq[i0] : 0.0f;
            int i1 = start + t + 1; i1 = (i1 > T_LEN - 1) ? (T_LEN - 1) : i1;
            v.y = (t + 1 < take) ? ldsSeq[i1] : 0.0f;
            int i2 = start + t + 2; i2 = (i2 > T_LEN - 1) ? (T_LEN - 1) : i2;
            v.z = (t + 2 < take) ? ldsSeq[i2] : 0.0f;
            int i3 = start + t + 3; i3 = (i3 > T_LEN - 1) ? (T_LEN - 1) : i3;
            v.w = (t + 3 < take) ? ldsSeq[i3] : 0.0f;
            out4[t4] = v;
        }
    }
}

extern "C" void kernel_launch(void* const* d_in, const int* in_sizes, int n_in,
                              void* d_out, int out_size, void* d_ws, size_t ws_size,
                              hipStream_t stream) {
    (void)in_sizes; (void)n_in; (void)out_size; (void)d_ws; (void)ws_size;
    const float* x = (const float*)d_in[0];
    float* out = (float*)d_out;
    periodicity_kernel<<<dim3(BN_TOTAL), dim3(256), 0, stream>>>(x, out);
}